// Motion_BiWKV_62294205661368
// MI455X (gfx1250) — compile-verified
//
#include <hip/hip_runtime.h>
#include <hip/hip_bf16.h>
#include <stdint.h>

// Problem constants (from reference): B=2, T=512, C=256
#define B_   2
#define T_   512
#define C_   256
#define EPS_ 1e-6f

#define CT   32            // channels per block (one lane per channel)
#define ROWS 8             // waves per block
#define TPT  4             // t-values per thread
#define TB   (ROWS * TPT)  // 32 t per block

typedef __attribute__((ext_vector_type(4))) unsigned int u32x4;
typedef __attribute__((ext_vector_type(8))) int          i32x8;
typedef __attribute__((ext_vector_type(4))) int          i32x4;

#if defined(__gfx1250__) && __has_builtin(__builtin_amdgcn_tensor_load_to_lds) && \
    __has_builtin(__builtin_amdgcn_s_wait_tensorcnt)
#define USE_TDM 1
#else
#define USE_TDM 0
#endif

__global__ __launch_bounds__(256)
void biwkv_kernel(const float* __restrict__ r,
                  const float* __restrict__ k,
                  const float* __restrict__ v,
                  const float* __restrict__ w,
                  float* __restrict__ out)
{
    __shared__ float vs[T_][CT];   // 64 KB: v[b, 0..T-1, c0..c0+31]

    const int tid  = threadIdx.x;
    const int lane = tid & 31;
    const int row  = tid >> 5;          // wave id within block
    const int b    = blockIdx.z;
    const int c0   = blockIdx.y * CT;
    const int t0   = blockIdx.x * TB;
    const int c    = c0 + lane;

    const float* vslice = v + (size_t)b * T_ * C_ + c0;

#if USE_TDM
    // ---- Tensor Data Mover: DMA the 2D tile (T_ rows x CT f32, stride C_)
    //      from global into LDS with one instruction (wave 0 only).
    if (row == 0) {
        const uint64_t gaddr = (uint64_t)(uintptr_t)vslice;
        const uint32_t laddr = (uint32_t)(uintptr_t)&vs[0][0]; // low 32b = LDS offset

        u32x4 g0;
        g0.x = 1u;                                   // count=1 (valid user descriptor)
        g0.y = laddr;                                // lds_addr  [63:32]
        g0.z = (uint32_t)gaddr;                      // global_addr[31:0]
        g0.w = (uint32_t)((gaddr >> 32) & 0x01FFFFFFu) | 0x80000000u; // addr[56:32] | type=2

        i32x8 g1;
        g1[0] = (2 << 16);                           // data_size=4B; mask/flags/pad = 0
        g1[1] = (int)((CT & 0xFFFF) << 16);          // tensor_dim0[15:0] = 32
        g1[2] = (int)((T_ & 0xFFFF) << 16);          // tensor_dim0 hi=0 | tensor_dim1[15:0]=512
        g1[3] = (int)((CT & 0xFFFF) << 16);          // tensor_dim1 hi=0 | tile_dim0=32
        g1[4] = (int)(T_ & 0xFFFF);                  // tile_dim1=512 | tile_dim2=0
        g1[5] = (int)C_;                             // tensor_dim0_stride[31:0] = 256 elems
        g1[6] = 0;                                   // stride0 hi | stride1 lo (2D: unused)
        g1[7] = 0;                                   // stride1 hi

        i32x4 z4 = {0, 0, 0, 0};                     // groups 2/3 unused (2D tensor)
        i32x8 z8 = {0, 0, 0, 0, 0, 0, 0, 0};         // extra operand (clang-23 6-arg form)
        __builtin_amdgcn_tensor_load_to_lds(g0, g1, z4, z4, z8, 0);
        __builtin_amdgcn_s_wait_tensorcnt(0);
    }
    __syncthreads();
#else
    // Fallback: coalesced vector-load staging (128B per wave per row)
    for (int i = tid; i < T_ * CT; i += 256) {
        int l = i >> 5, cc = i & 31;
        vs[l][cc] = vslice[(size_t)l * C_ + cc];
    }
    __syncthreads();
#endif

    // Per-channel decay: wd[c] = -exp(w[c])
    const float wd  = -__expf(w[c]);
    const float inv = 1.0f / (float)(T_ - 1);

    float num[TPT], den[TPT], ratio[TPT], m0[TPT], beta[TPT], sg[TPT];

#pragma unroll
    for (int j = 0; j < TPT; ++j) {
        const int t = t0 + row * TPT + j;
        const size_t idx = ((size_t)(b * T_ + t)) * C_ + c;
        const float kk = k[idx];
        const float rr = r[idx];
        sg[j]   = 1.0f / (1.0f + __expf(-rr));   // sigmoid(r)
        beta[j] = kk * wd * inv;                  // score slope per l
        const float s0 = beta[j] * (float)(T_ - 1);
        m0[j]   = fminf(s0, 0.0f);               // exact max-subtraction constant
        ratio[j] = __expf(-beta[j]);             // geometric ratio a_{l+1}/a_l
        num[j] = 0.0f;
        den[j] = 0.0f;
    }

    // Softmax-weighted sum over l via geometric recurrence:
    //   a_l = exp(m0 - beta*l);  a_{l+1} = a_l * ratio
    // Refresh exactly every 16 steps to bound rounding drift.
    float p[TPT];
    for (int l0 = 0; l0 < T_; l0 += 16) {
#pragma unroll
        for (int j = 0; j < TPT; ++j)
            p[j] = __expf(m0[j] - beta[j] * (float)l0);
#pragma unroll
        for (int li = 0; li < 16; ++li) {
            const float vv = vs[l0 + li][lane];   // bank-conflict-free broadcast row
#pragma unroll
            for (int j = 0; j < TPT; ++j) {
                num[j] = fmaf(p[j], vv, num[j]);
                den[j] += p[j];
                p[j]   *= ratio[j];
            }
        }
    }

    // x_b == x_f exactly (index relabeling), so out = sigmoid(r) * x_f.
#pragma unroll
    for (int j = 0; j < TPT; ++j) {
        const int t = t0 + row * TPT + j;
        const size_t idx = ((size_t)(b * T_ + t)) * C_ + c;
        out[idx] = sg[j] * num[j] / (den[j] + EPS_);
    }
}

extern "C" void kernel_launch(void* const* d_in, const int* in_sizes, int n_in,
                              void* d_out, int out_size, void* d_ws, size_t ws_size,
                              hipStream_t stream) {
    const float* r = (const float*)d_in[0];
    const float* k = (const float*)d_in[1];
    const float* v = (const float*)d_in[2];
    const float* w = (const float*)d_in[3];
    // d_in[4] (u) is unused by the reference computation.
    float* out = (float*)d_out;

    dim3 grid(T_ / TB, C_ / CT, B_);   // (16, 8, 2) = 256 blocks
    biwkv_kernel<<<grid, dim3(256), 0, stream>>>(r, k, v, w, out);
    (void)in_sizes; (void)n_in; (void)out_size; (void)d_ws; (void)ws_size;
}